// ValleySeparationLoss_81346680586854
// MI455X (gfx1250) — compile-verified
//
#include <hip/hip_runtime.h>
#include <hip/hip_bf16.h>

typedef __attribute__((ext_vector_type(2))) float v2f;
typedef __attribute__((ext_vector_type(8))) float v8f;

#define Bimg 8
#define Hdim 128
#define Wdim 128
#define Cdim 256
#define Ncen 16
#define PIX  (Hdim * Wdim)          // 16384 pixels per image
#define TILES_PER_IMG (PIX / 16)    // 1024 tiles of 16 pixels
#define WAVES_PER_BLOCK 4

// workspace layout (floats)
#define WS_GT     0                         // B*16*256 = 32768 floats (gt feats, row-major per center)
#define WS_GNORM  (WS_GT + Bimg*Ncen*Cdim)  // B*16 floats (+pad)
#define WS_CMAP   (WS_GNORM + 256)          // B*16384 floats

// ---------------------------------------------------------------------------
// Kernel A: gather gt features (transposed to per-center contiguous rows) and
// compute per-center squared norms. One block per image.
// ---------------------------------------------------------------------------
__global__ __launch_bounds__(256)
void vs_gather_gt(const float* __restrict__ feat, const int* __restrict__ centers,
                  float* __restrict__ ws) {
    __shared__ float lgt[Ncen * Cdim];      // 16 KB
    const int img = blockIdx.x;
    const int t   = threadIdx.x;            // 0..255 == channel
    float* gtW    = ws + WS_GT + img * (Ncen * Cdim);
    float* gnormW = ws + WS_GNORM + img * Ncen;

    #pragma unroll
    for (int n = 0; n < Ncen; ++n) {
        const int r = centers[(img * Ncen + n) * 2 + 0];
        const int c = centers[(img * Ncen + n) * 2 + 1];
        const float v = feat[(((size_t)img * PIX) + (size_t)r * Wdim + c) * Cdim + t];
        lgt[n * Cdim + t] = v;
        gtW[n * Cdim + t] = v;              // staged for WMMA B-fragment loads
    }
    __syncthreads();
    if (t < Ncen) {
        float s = 0.f;
        #pragma unroll 8
        for (int k = 0; k < Cdim; ++k) { float v = lgt[t * Cdim + k]; s += v * v; }
        gnormW[t] = s;
    }
}

// ---------------------------------------------------------------------------
// Kernel B: per 16-pixel tile, WMMA f32 16x16x4 over K=256 to get pixel·center
// dot products; fuse |p|^2 accumulation; d2 = |p|^2 + |g|^2 - 2*dot;
// cmap = min_n sqrt(max(d2,0)). One wave per tile, 4 waves per block.
// ---------------------------------------------------------------------------
__global__ __launch_bounds__(WAVES_PER_BLOCK * 32)
void vs_cmap(const float* __restrict__ feat, const float* __restrict__ ws_ro,
             float* __restrict__ ws) {
    const int lane = threadIdx.x & 31;
    const int wave = threadIdx.x >> 5;
    const int tile = blockIdx.x * WAVES_PER_BLOCK + wave;
    const int img  = tile / TILES_PER_IMG;
    const int tIn  = tile - img * TILES_PER_IMG;
    const int pixBase = tIn * 16;

    const int mrow = lane & 15;             // A: pixel row; B: center column
    const int koff = (lane >> 4) * 2;       // fp32 WMMA half-wave K offset

    // A: this lane's pixel feature row; B: this lane's center feature row
    const float* aPtr = feat + (((size_t)img * PIX) + pixBase + mrow) * Cdim + koff;
    const float* bPtr = ws_ro + WS_GT + (size_t)img * (Ncen * Cdim) + mrow * Cdim + koff;

    v8f acc = {};                           // 16x16 f32 dot-product accumulator
    float pn = 0.f;                         // partial |p|^2 (this lane's k-subset)

    #pragma unroll 8
    for (int s = 0; s < Cdim / 4; ++s) {
        v2f a = *(const v2f*)(aPtr + 4 * s);
        v2f b = *(const v2f*)(bPtr + 4 * s);
        pn += a.x * a.x + a.y * a.y;
        acc = __builtin_amdgcn_wmma_f32_16x16x4_f32(
            /*neg_a=*/false, a, /*neg_b=*/false, b,
            /*c_mod=*/(short)0, acc, /*reuse_a=*/false, /*reuse_b=*/false);
    }

    // full |p|^2 for pixel row (lane&15): combine the two half-wave k-subsets
    const float pnF = pn + __shfl_xor(pn, 16, 32);
    const float gn  = ws_ro[WS_GNORM + img * Ncen + mrow];  // |g_n|^2, n = lane&15

    const int hi = lane >> 4;
    const int lq = lane & 7;
    float outv = 0.f;

    #pragma unroll
    for (int r = 0; r < 8; ++r) {
        // C/D layout: VGPR r holds row M = r + 8*hi, col N = lane&15
        const float pm = __shfl(pnF, r + 8 * hi, 32);
        float d2 = pm + gn - 2.f * acc[r];
        float dv = __builtin_sqrtf(fmaxf(d2, 0.f));
        // min over the 16 centers = min across each 16-lane half
        dv = fminf(dv, __shfl_xor(dv, 1, 32));
        dv = fminf(dv, __shfl_xor(dv, 2, 32));
        dv = fminf(dv, __shfl_xor(dv, 4, 32));
        dv = fminf(dv, __shfl_xor(dv, 8, 32));
        outv = (lq == r) ? dv : outv;
    }
    // lanes 0..7 hold rows 0..7 (lower half), lanes 24..31 hold rows 8..15
    if (lane < 8 || lane >= 24) {
        const int row = (lane < 8) ? lane : (lane - 16);
        ws[WS_CMAP + (size_t)img * PIX + pixBase + row] = outv;
    }
}

// ---------------------------------------------------------------------------
// Kernel C: valley loss over all B*120 pairs, single block reduction.
// ---------------------------------------------------------------------------
__global__ __launch_bounds__(1024)
void vs_valley(const float* __restrict__ ws, const int* __restrict__ centers,
               float* __restrict__ out) {
    __shared__ float red[1024];
    const int t = threadIdx.x;
    float val = 0.f;
    const int NP = Ncen * (Ncen - 1) / 2;   // 120
    if (t < Bimg * NP) {
        const int img = t / NP;
        int q = t - img * NP;
        int i = 0;
        while (q >= (Ncen - 1 - i)) { q -= (Ncen - 1 - i); ++i; }
        const int j = i + 1 + q;
        const int r1 = centers[(img * Ncen + i) * 2 + 0];
        const int c1 = centers[(img * Ncen + i) * 2 + 1];
        const int r2 = centers[(img * Ncen + j) * 2 + 0];
        const int c2 = centers[(img * Ncen + j) * 2 + 1];
        const float* cmap = ws + WS_CMAP + (size_t)img * PIX;
        const float e1 = cmap[r1 * Wdim + c1];
        const float e2 = cmap[r2 * Wdim + c2];
        // t_mid = 0.5; jnp.round == round-half-to-even == rintf
        const int rm = (int)rintf((float)r1 + 0.5f * (float)(r2 - r1));
        const int cm = (int)rintf((float)c1 + 0.5f * (float)(c2 - c1));
        const float mid = cmap[rm * Wdim + cm];
        val = fmaxf(0.5f * (e1 + e2) * 1.2f - mid, 0.f);
    }
    red[t] = val;
    __syncthreads();
    for (int s = 512; s > 0; s >>= 1) {
        if (t < s) red[t] += red[t + s];
        __syncthreads();
    }
    if (t == 0) out[0] = red[0] / (float)(Bimg * NP);
}

extern "C" void kernel_launch(void* const* d_in, const int* in_sizes, int n_in,
                              void* d_out, int out_size, void* d_ws, size_t ws_size,
                              hipStream_t stream) {
    const float* feat    = (const float*)d_in[0];
    const int*   centers = (const int*)d_in[1];
    float* ws  = (float*)d_ws;
    float* out = (float*)d_out;

    vs_gather_gt<<<Bimg, 256, 0, stream>>>(feat, centers, ws);
    const int totalTiles = Bimg * TILES_PER_IMG;                 // 8192 waves
    vs_cmap<<<totalTiles / WAVES_PER_BLOCK, WAVES_PER_BLOCK * 32, 0, stream>>>(feat, ws, ws);
    vs_valley<<<1, 1024, 0, stream>>>(ws, centers, out);
}